// AdSBHNet_50654844289325
// MI455X (gfx1250) — compile-verified
//
#include <hip/hip_runtime.h>
#include <math.h>

#define NPTS 1000
#define NEWTON_ITERS 8
#define BISECT_ITERS 40
#define NGRID 256
#define TPB 256
#define NW (TPB / 32)
#define PPT 4

static __device__ const float F_PI = 3.14159265358979f;
static __device__ const float F_EPS = 1e-3f;

typedef __attribute__((ext_vector_type(2))) float v2f;
typedef __attribute__((ext_vector_type(8))) float v8f;

struct C2 { float x, y; };

__device__ inline C2 mkc(float x, float y) { C2 r; r.x = x; r.y = y; return r; }
__device__ inline C2 cadd(C2 a, C2 b) { return mkc(a.x + b.x, a.y + b.y); }
__device__ inline C2 csub(C2 a, C2 b) { return mkc(a.x - b.x, a.y - b.y); }
__device__ inline C2 cmul(C2 a, C2 b) { return mkc(a.x * b.x - a.y * b.y, a.x * b.y + a.y * b.x); }
__device__ inline C2 crmul(C2 a, float s) { return mkc(a.x * s, a.y * s); }
__device__ inline C2 cdiv(C2 a, C2 b) {
  float d = b.x * b.x + b.y * b.y;
  float inv = 1.0f / d;
  return mkc((a.x * b.x + a.y * b.y) * inv, (a.y * b.x - a.x * b.y) * inv);
}
__device__ inline C2 cinv(C2 b) {
  float d = b.x * b.x + b.y * b.y;
  float inv = 1.0f / d;
  return mkc(b.x * inv, -b.y * inv);
}
__device__ inline C2 clogC(C2 z) {
  return mkc(0.5f * logf(z.x * z.x + z.y * z.y), atan2f(z.y, z.x));
}
__device__ inline C2 csqrtC(C2 z) {
  float m = sqrtf(z.x * z.x + z.y * z.y);
  float u = sqrtf(0.5f * (m + fabsf(z.x)));
  C2 r;
  if (z.x >= 0.0f) {
    r.x = u;
    r.y = (u > 0.0f) ? (0.5f * z.y / u) : 0.0f;
  } else {
    r.x = (u > 0.0f) ? (0.5f * fabsf(z.y) / u) : 0.0f;
    r.y = copysignf(u, z.y);
  }
  return r;
}
__device__ inline C2 cpow15(C2 w) { return cmul(w, csqrtC(w)); }

// ---------------- wave32 sum via V_WMMA_F32_16X16X4_F32 (ones B-matrix) ----
// A layout (16x4 f32): lane L<16 holds A[L][0],A[L][1]; lane L>=16 holds
// A[L-16][2],A[L-16][3]. With B==ones, D[m][n] = sum_k A[m][k], so lane n<16
// gets rows 0..7 in its 8 D VGPRs and lane n>=16 gets rows 8..15; summing the
// 8 VGPRs per lane and adding across the half-wave boundary yields the full
// 32-lane sum in every lane.
__device__ inline float wave_sum(float x) {
#if __has_builtin(__builtin_amdgcn_wmma_f32_16x16x4_f32)
  v2f A; A.x = x; A.y = 0.0f;
  v2f B; B.x = 1.0f; B.y = 1.0f;
  v8f Cz = {0.f, 0.f, 0.f, 0.f, 0.f, 0.f, 0.f, 0.f};
  v8f D = __builtin_amdgcn_wmma_f32_16x16x4_f32(false, A, false, B, (short)0,
                                                Cz, false, false);
  float s = D[0] + D[1] + D[2] + D[3] + D[4] + D[5] + D[6] + D[7];
  s += __shfl_xor(s, 16, 32);
  return s;
#else
  float s = x;
  for (int o = 16; o >= 1; o >>= 1) s += __shfl_xor(s, o, 32);
  return s;
#endif
}

__device__ inline C2 block_sum(C2 v, float* lds) {
  float sr = wave_sum(v.x);
  float si = wave_sum(v.y);
  int lane = threadIdx.x & 31;
  int wv = threadIdx.x >> 5;
  __syncthreads();
  if (lane == 0) { lds[2 * wv] = sr; lds[2 * wv + 1] = si; }
  __syncthreads();
  C2 out = mkc(0.f, 0.f);
  #pragma unroll
  for (int w = 0; w < NW; ++w) { out.x += lds[2 * w]; out.y += lds[2 * w + 1]; }
  return out;
}

// ---------------- model evaluation -----------------------------------------
__device__ inline float compute_coefs(const float* ga, const float* gb,
                                      float* c, float* bb) {
  float _a[6];
  _a[0] = 1.f;
  #pragma unroll
  for (int i = 0; i < 5; ++i) _a[i + 1] = ga[i];
  #pragma unroll
  for (int k = 0; k < 11; ++k) c[k] = 0.f;
  #pragma unroll
  for (int i = 0; i < 6; ++i)
    #pragma unroll
    for (int j = 0; j < 6; ++j) c[i + j] += 4.f * _a[i] * _a[j];
  bb[0] = 1.f;
  #pragma unroll
  for (int i = 0; i < 5; ++i) bb[i + 1] = gb[i];
  float sa2 = 0.f;
  #pragma unroll
  for (int i = 0; i < 5; ++i) sa2 += ga[i] * ga[i];
  return sa2;
}

__device__ inline void eval_all(C2 z, const float* c, float sa2,
                                const float* bb, C2& f, C2& df, C2& bv,
                                C2& dbv) {
  C2 zp[11];
  zp[0] = mkc(1.f, 0.f);
  #pragma unroll
  for (int k = 1; k < 11; ++k) zp[k] = cmul(zp[k - 1], z);
  C2 lz = clogC(z);
  C2 acc = mkc(0.f, 0.f), sck = mkc(0.f, 0.f);
  #pragma unroll
  for (int k = 0; k < 11; ++k) {
    sck.x += c[k] * zp[k].x;
    sck.y += c[k] * zp[k].y;
    if (k != 4) {
      float s = c[k] / (float)(k - 4);
      acc.x += s * (zp[4].x - zp[k].x);
      acc.y += s * (zp[4].y - zp[k].y);
    }
  }
  C2 z4lz = cmul(zp[4], lz);
  acc.x -= c[4] * z4lz.x;
  acc.y -= c[4] * z4lz.y;
  float e4 = 4.f * F_EPS * sa2;
  C2 omz = mkc(1.f - z.x, -z.y);
  C2 z4omz = cmul(zp[4], omz);
  f = mkc(acc.x + e4 * z4omz.x, acc.y + e4 * z4omz.y);
  C2 tmp = mkc(4.f * f.x - sck.x, 4.f * f.y - sck.y);
  C2 dfz = cdiv(tmp, z);
  df = mkc(dfz.x - e4 * zp[4].x, dfz.y - e4 * zp[4].y);
  bv = mkc(0.f, 0.f);
  dbv = mkc(0.f, 0.f);
  #pragma unroll
  for (int i = 0; i < 6; ++i) {
    bv.x += bb[i] * zp[i].x;
    bv.y += bb[i] * zp[i].y;
    if (i >= 1) {
      dbv.x += (float)i * bb[i] * zp[i - 1].x;
      dbv.y += (float)i * bb[i] * zp[i - 1].y;
    }
  }
}

// _extrap_trapz weights folded analytically into per-point weights.
__device__ inline float trapwY(int p, float h) {
  const float Y0 = 1e-3f;
  if (p == 0) return 0.5f * h + Y0 + 0.5f * Y0 * Y0 / h;
  if (p == 1) return h - 0.5f * Y0 * Y0 / h;
  if (p == NPTS - 1) return 0.5f * h + 0.5f * (1.f - 0.999f);
  return h;
}

// cooperative integrate_L + integrate_dL at one (complex) zs
__device__ inline void integrals_LdL(C2 zs, const float* c, float sa2,
                                     const float* bb, float* lds, C2& Lout,
                                     C2& dLout) {
  C2 fs, dfs, bvs, dbvs;
  eval_all(zs, c, sa2, bb, fs, dfs, bvs, dbvs);
  C2 zdfs_fs = cmul(zs, cdiv(dfs, fs));
  C2 accL = mkc(0.f, 0.f), accD = mkc(0.f, 0.f);
  const float h = (0.999f - 1e-3f) / 999.f;
  const C2 one = mkc(1.f, 0.f);
  for (int q = 0; q < PPT; ++q) {
    int p = threadIdx.x + TPB * q;
    if (p < NPTS) {
      float Y = 1e-3f + h * (float)p;
      float wt = trapwY(p, h);
      float w = 1.f - Y * Y;
      float w2 = w * w, w4 = w2 * w2;
      float r4 = 1.f / w4;
      C2 z = mkc(zs.x * w, zs.y * w);
      C2 f, df, bv, dbv;
      eval_all(z, c, sa2, bb, f, df, bv, dbv);
      C2 g = cdiv(cmul(bv, bv), f);
      C2 dg = cdiv(csub(crmul(cmul(bv, dbv), 2.f),
                        cmul(cmul(bv, bv), cdiv(df, f))), f);
      C2 fofs = cdiv(f, fs);
      C2 sg = csqrtC(g);
      C2 t1 = mkc(fofs.x * r4 - 1.f, fofs.y * r4);  // fofs/w4 - 1
      // L integrand
      C2 iL = cdiv(sg, csqrtC(t1));
      accL.x += iL.x * (Y * wt);
      accL.y += iL.y * (Y * wt);
      // dL integrand
      C2 zdg_g = cmul(z, cdiv(dg, g));
      C2 A1 = mkc(zdfs_fs.x + 2.f + zdg_g.x, zdfs_fs.y + zdg_g.y);
      C2 dfofs = cdiv(df, fs);
      C2 term = csub(csub(crmul(cmul(fofs, A1), r4),
                          crmul(cmul(z, dfofs), r4)),
                     mkc(2.f + zdg_g.x, zdg_g.y));
      C2 sq1 = csqrtC(csub(one, cdiv(z, zs)));  // == Y up to rounding
      C2 iD = cdiv(crmul(cmul(cmul(term, sq1), sg), 2.f), cpow15(t1));
      accD.x += iD.x * wt;
      accD.y += iD.y * wt;
    }
  }
  C2 TL = block_sum(accL, lds);
  C2 TD = block_sum(accD, lds);
  Lout = crmul(cmul(zs, TL), 4.f / F_PI);
  dLout = crmul(TD, 1.f / F_PI);
}

// ---------------- kernels ---------------------------------------------------
__global__ void k_bisect(const float* ga, const float* gb, float* ws) {
  __shared__ float lds[2 * NW];
  float c[11], bb[6];
  float sa2 = compute_coefs(ga, gb, c, bb);
  float lo = 1e-3f, hi = 0.999f;
  for (int it = 0; it < BISECT_ITERS; ++it) {
    float mid = 0.5f * (lo + hi);
    C2 Lv, dLv;
    integrals_LdL(mkc(mid, 0.f), c, sa2, bb, lds, Lv, dLv);
    if (dLv.x < 0.f) hi = mid; else lo = mid;  // uniform after reduction
  }
  if (threadIdx.x == 0) ws[0] = 0.5f * (lo + hi);
}

__global__ void k_grid(const float* ga, const float* gb, float* ws) {
  __shared__ float lds[2 * NW];
  float c[11], bb[6];
  float sa2 = compute_coefs(ga, gb, c, bb);
  float zs_max = ws[0];
  float t = 0.05f + (0.999f - 0.05f) * (float)blockIdx.x / (float)(NGRID - 1);
  C2 Lv, dLv;
  integrals_LdL(mkc(zs_max * t, 0.f), c, sa2, bb, lds, Lv, dLv);
  if (threadIdx.x == 0) ws[16 + blockIdx.x] = Lv.x;
}

__device__ inline C2 calc_Vc(C2 zs, const float* c, float sa2, const float* bb,
                             float* lds, float expc) {
  C2 fs, dfs, bvs, dbvs;
  eval_all(zs, c, sa2, bb, fs, dfs, bvs, dbvs);
  C2 acc = mkc(0.f, 0.f);
  const float h = (0.999f - 1e-3f) / 999.f;
  const C2 one = mkc(1.f, 0.f);
  for (int q = 0; q < PPT; ++q) {
    int p = threadIdx.x + TPB * q;
    if (p < NPTS) {
      float Y = 1e-3f + h * (float)p;
      float wt = trapwY(p, h);
      float w = 1.f - Y * Y;
      float w2 = w * w, w4 = w2 * w2;
      C2 z = mkc(zs.x * w, zs.y * w);
      C2 f, df, bv, dbv;
      eval_all(z, c, sa2, bb, f, df, bv, dbv);
      C2 g = cdiv(cmul(bv, bv), f);
      C2 fg = cmul(f, g);
      C2 fofs = cdiv(f, fs);
      C2 t = csub(one, crmul(cinv(fofs), w4));  // 1 - w2^2/fofs
      C2 inner = csub(cdiv(one, csqrtC(t)), one);
      C2 sfg = csqrtC(fg);
      C2 iV = cmul(mkc(sfg.x / w2, sfg.y / w2), inner);
      acc.x += iV.x * (Y * wt);
      acc.y += iV.y * (Y * wt);
    }
  }
  C2 T = block_sum(acc, lds);
  return cdiv(crmul(T, expc * F_PI * 4.f), zs);
}

__device__ inline C2 calc_Vd(C2 zs, const float* c, float sa2, const float* bb,
                             float* lds, float expc) {
  C2 omzs = mkc(1.f - zs.x, -zs.y);
  const float hd = (1.f - 1e-3f) / 999.f;
  C2 acc = mkc(0.f, 0.f);
  for (int q = 0; q < PPT; ++q) {
    int p = threadIdx.x + TPB * q;
    if (p < NPTS) {
      float yd = 1e-3f + hd * (float)p;
      float wt = (p == 0) ? 0.5f * (1e-3f + hd)
                          : ((p == NPTS - 1) ? 0.5f * hd : hd);
      C2 z = mkc(1.f - omzs.x * yd, -omzs.y * yd);
      C2 f, df, bv, dbv;
      eval_all(z, c, sa2, bb, f, df, bv, dbv);
      C2 g = cdiv(cmul(bv, bv), f);
      C2 iV = cdiv(csqrtC(cmul(f, g)), cmul(z, z));
      acc.x += iV.x * wt;
      acc.y += iV.y * wt;
    }
  }
  C2 T = block_sum(acc, lds);
  T.x += 0.5f * 1e-3f;  // prepended f(0)=1 trapezoid segment
  return crmul(cmul(omzs, T), expc * F_PI * 2.f);
}

__global__ void k_main(const float* Ls, const float* ga, const float* gb,
                       const float* lc, float* out, const float* ws, int cplx) {
  __shared__ float lds[2 * NW];
  float c[11], bb[6];
  float sa2 = compute_coefs(ga, gb, c, bb);
  float Ltgt = Ls[blockIdx.x];
  float expc = expf(lc[0]);
  float zs_max = ws[0];
  const float* Lg = ws + 16;
  // argmin |Lg - L| (uniform scan; ties -> first, matching jnp.argmin)
  int best = 0;
  float bd = fabsf(Lg[0] - Ltgt);
  for (int i = 1; i < NGRID; ++i) {
    float d = fabsf(Lg[i] - Ltgt);
    if (d < bd) { bd = d; best = i; }
  }
  C2 zs = mkc(zs_max * (0.05f + (0.999f - 0.05f) * (float)best /
                                  (float)(NGRID - 1)),
              0.f);
  for (int it = 0; it < NEWTON_ITERS; ++it) {
    C2 Lv, dLv;
    integrals_LdL(zs, c, sa2, bb, lds, Lv, dLv);
    C2 stp = cdiv(mkc(Lv.x - Ltgt, Lv.y), dLv);
    zs.x -= stp.x;
    zs.y -= stp.y;
  }
  zs.y = fabsf(zs.y);
  C2 Vc = calc_Vc(zs, c, sa2, bb, lds, expc);
  C2 Vd = calc_Vd(zs, c, sa2, bb, lds, expc);
  C2 o = csub(Vc, Vd);
  if (threadIdx.x == 0) {
    if (cplx) {
      out[2 * blockIdx.x] = o.x;
      out[2 * blockIdx.x + 1] = o.y;
    } else {
      out[blockIdx.x] = o.x;
    }
  }
}

extern "C" void kernel_launch(void* const* d_in, const int* in_sizes, int n_in,
                              void* d_out, int out_size, void* d_ws,
                              size_t ws_size, hipStream_t stream) {
  const float* Ls = (const float*)d_in[0];
  const float* a = (const float*)d_in[1];
  const float* b = (const float*)d_in[2];
  const float* lc = (const float*)d_in[3];
  int B = in_sizes[0];
  float* ws = (float*)d_ws;
  float* out = (float*)d_out;
  int cplx = (out_size >= 2 * B) ? 1 : 0;
  k_bisect<<<1, TPB, 0, stream>>>(a, b, ws);
  k_grid<<<NGRID, TPB, 0, stream>>>(a, b, ws);
  k_main<<<B, TPB, 0, stream>>>(Ls, a, b, lc, out, ws, cplx);
}